// SAGE_59846074302982
// MI455X (gfx1250) — compile-verified
//
#include <hip/hip_runtime.h>
#include <hip/hip_bf16.h>

#define NNODES 50000
#define NEDGES 800000
#define NHID   128
#define NCLASS 8

typedef __attribute__((ext_vector_type(2))) float v2f;
typedef __attribute__((ext_vector_type(8))) float v8f;

// ---------------------------------------------------------------------------
// Degree: one float atomic per edge
__global__ __launch_bounds__(256) void deg_kernel(const long long* __restrict__ dst,
                                                  float* __restrict__ deg) {
    int e = blockIdx.x * blockDim.x + threadIdx.x;
    if (e < NEDGES) atomicAdd(&deg[(int)dst[e]], 1.0f);
}

__global__ __launch_bounds__(256) void invdeg_kernel(const float* __restrict__ deg,
                                                     float* __restrict__ invdeg) {
    int i = blockIdx.x * blockDim.x + threadIdx.x;
    if (i < NNODES) {
        float d = deg[i];
        invdeg[i] = (d > 0.0f) ? (1.0f / d) : 0.0f;
    }
}

// ---------------------------------------------------------------------------
// Scatter-add of neighbor features: one wave per edge, float4 per lane.
__global__ __launch_bounds__(256) void scatter_kernel(const long long* __restrict__ src,
                                                      const long long* __restrict__ dst,
                                                      const float* __restrict__ X,
                                                      float* __restrict__ Agg) {
    int e = blockIdx.x * 8 + (threadIdx.x >> 5);
    if (e >= NEDGES) return;
    int lane = threadIdx.x & 31;
    int s = (int)src[e];
    int d = (int)dst[e];
    float4 v = ((const float4*)(X + (size_t)s * NHID))[lane];
    float* out = Agg + (size_t)d * NHID + lane * 4;
    atomicAdd(out + 0, v.x);
    atomicAdd(out + 1, v.y);
    atomicAdd(out + 2, v.z);
    atomicAdd(out + 3, v.w);
}

// ---------------------------------------------------------------------------
// Fused SAGE GEMM: Out = (Agg * invdeg) @ Wl^T + bl + X @ Wr^T
// One wave per 16x16 output tile using V_WMMA_F32_16X16X4_F32 (64 WMMAs/tile).
// Block = 8 waves covering the 8 column tiles of one 16-row block (A reuse in L1).
__global__ __launch_bounds__(256) void sage_gemm_kernel(const float* __restrict__ Agg,
                                                        const float* __restrict__ invdeg,
                                                        const float* __restrict__ X,
                                                        const float* __restrict__ Wl,
                                                        const float* __restrict__ bl,
                                                        const float* __restrict__ Wr,
                                                        float* __restrict__ Out) {
    const int lane  = threadIdx.x & 31;
    const int wave  = threadIdx.x >> 5;        // column tile 0..7
    const int r0    = blockIdx.x * 16;         // row block (3125 * 16 == 50000)
    const int c0    = wave * 16;
    const int m     = lane & 15;               // M (A) and N (B) index for this lane
    const int khalf = (lane >> 4) << 1;        // lanes 16..31 hold K+2,K+3

    const int   row = r0 + m;
    const float sA  = invdeg[row];

    const float* arow  = Agg + (size_t)row * NHID + khalf;
    const float* xrow  = X   + (size_t)row * NHID + khalf;
    const float* wlrow = Wl  + (size_t)(c0 + m) * NHID + khalf; // B[k][n] = Wl[n][k]
    const float* wrrow = Wr  + (size_t)(c0 + m) * NHID + khalf;

    v8f acc = {};
    // Pass 1: mean-aggregated neighbor branch through Wl (row-scaled by invdeg)
    for (int k0 = 0; k0 < NHID; k0 += 4) {
        v2f a = *(const v2f*)(arow + k0);
        a = a * sA;
        v2f b = *(const v2f*)(wlrow + k0);
        acc = __builtin_amdgcn_wmma_f32_16x16x4_f32(false, a, false, b,
                                                    (short)0, acc, false, false);
    }
    // Pass 2: root branch through Wr
    for (int k0 = 0; k0 < NHID; k0 += 4) {
        v2f a = *(const v2f*)(xrow + k0);
        v2f b = *(const v2f*)(wrrow + k0);
        acc = __builtin_amdgcn_wmma_f32_16x16x4_f32(false, a, false, b,
                                                    (short)0, acc, false, false);
    }

    // D layout: VGPR v -> M = v + 8*(lane>=16), N = lane&15
    const float bias      = bl[c0 + m];
    const int   mrow_base = r0 + ((lane >> 4) << 3);
    for (int v = 0; v < 8; ++v) {
        Out[(size_t)(mrow_base + v) * NHID + c0 + m] = acc[v] + bias;
    }
}

// ---------------------------------------------------------------------------
// Per-row L2 normalize (+ optional ReLU). One wave per row, float4 per lane.
__global__ __launch_bounds__(256) void rownorm_kernel(float* __restrict__ H, int do_relu) {
    int row  = (blockIdx.x * blockDim.x + threadIdx.x) >> 5;
    int lane = threadIdx.x & 31;
    if (row >= NNODES) return;
    float4* p = (float4*)(H + (size_t)row * NHID);
    float4 v = p[lane];
    float ss = v.x * v.x + v.y * v.y + v.z * v.z + v.w * v.w;
    #pragma unroll
    for (int off = 16; off > 0; off >>= 1) ss += __shfl_xor(ss, off, 32);
    float inv = 1.0f / fmaxf(sqrtf(ss), 1e-12f);
    v.x *= inv; v.y *= inv; v.z *= inv; v.w *= inv;
    if (do_relu) {
        v.x = fmaxf(v.x, 0.0f); v.y = fmaxf(v.y, 0.0f);
        v.z = fmaxf(v.z, 0.0f); v.w = fmaxf(v.w, 0.0f);
    }
    p[lane] = v;
}

// ---------------------------------------------------------------------------
// Column sums / sums-of-squares: block-local accumulation, 2 atomics/col/block.
__global__ __launch_bounds__(128) void colstats_kernel(const float* __restrict__ H,
                                                       float* __restrict__ colsum,
                                                       float* __restrict__ colsq) {
    int t  = threadIdx.x;            // feature column
    int r0 = blockIdx.x * 128;
    int rmax = NNODES - r0; if (rmax > 128) rmax = 128;
    float s = 0.0f, q = 0.0f;
    for (int r = 0; r < rmax; ++r) {
        float v = H[(size_t)(r0 + r) * NHID + t];
        s += v; q += v * v;
    }
    atomicAdd(&colsum[t], s);
    atomicAdd(&colsq[t], q);
}

__global__ __launch_bounds__(128) void bn_finalize_kernel(const float* __restrict__ colsum,
                                                          const float* __restrict__ colsq,
                                                          const float* __restrict__ gamma,
                                                          const float* __restrict__ beta,
                                                          float* __restrict__ scale,
                                                          float* __restrict__ shift) {
    int t = threadIdx.x;
    float invn = 1.0f / (float)NNODES;
    float mean = colsum[t] * invn;
    float var  = colsq[t] * invn - mean * mean;
    float sc   = gamma[t] * rsqrtf(var + 1e-5f);
    scale[t] = sc;
    shift[t] = beta[t] - mean * sc;
}

__global__ __launch_bounds__(256) void bn_apply_kernel(float* __restrict__ H,
                                                       const float* __restrict__ scale,
                                                       const float* __restrict__ shift) {
    int i = blockIdx.x * blockDim.x + threadIdx.x;
    if (i < NNODES * NHID) {
        int f = i & (NHID - 1);
        H[i] = H[i] * scale[f] + shift[f];
    }
}

// ---------------------------------------------------------------------------
// Final FC: [N,128] @ [8,128]^T + b. Only 8 output cols -> scalar dot per thread.
__global__ __launch_bounds__(256) void fc_kernel(const float* __restrict__ H,
                                                 const float* __restrict__ Wfc,
                                                 const float* __restrict__ bfc,
                                                 float* __restrict__ Out) {
    int idx = blockIdx.x * blockDim.x + threadIdx.x;
    if (idx >= NNODES * NCLASS) return;
    int n = idx >> 3, c = idx & 7;
    const float4* h = (const float4*)(H + (size_t)n * NHID);
    const float4* w = (const float4*)(Wfc + (size_t)c * NHID);
    float s = 0.0f;
    #pragma unroll 8
    for (int i = 0; i < NHID / 4; ++i) {
        float4 a = h[i], b = w[i];
        s += a.x * b.x + a.y * b.y + a.z * b.z + a.w * b.w;
    }
    Out[idx] = s + bfc[c];
}

// ---------------------------------------------------------------------------
extern "C" void kernel_launch(void* const* d_in, const int* in_sizes, int n_in,
                              void* d_out, int out_size, void* d_ws, size_t ws_size,
                              hipStream_t stream) {
    const float*     x     = (const float*)d_in[0];
    const long long* ei    = (const long long*)d_in[1];   // int64 [2, E]
    const float*     W1l   = (const float*)d_in[2];
    const float*     b1l   = (const float*)d_in[3];
    const float*     W1r   = (const float*)d_in[4];
    const float*     gamma = (const float*)d_in[5];
    const float*     beta  = (const float*)d_in[6];
    const float*     W2l   = (const float*)d_in[7];
    const float*     b2l   = (const float*)d_in[8];
    const float*     W2r   = (const float*)d_in[9];
    const float*     Wfc   = (const float*)d_in[10];
    const float*     bfc   = (const float*)d_in[11];
    float*           out   = (float*)d_out;

    const long long* src = ei;
    const long long* dst = ei + NEDGES;

    // workspace layout (floats)
    float* ws     = (float*)d_ws;
    float* deg    = ws;                                   // [N]
    float* invdeg = deg + NNODES;                         // [N]
    float* agg    = invdeg + NNODES;                      // [N*128]
    float* h1     = agg + (size_t)NNODES * NHID;          // [N*128]
    float* h2     = h1 + (size_t)NNODES * NHID;           // [N*128]
    float* colsum = h2 + (size_t)NNODES * NHID;           // [128]
    float* colsq  = colsum + NHID;                        // [128]
    float* scale  = colsq + NHID;                         // [128]
    float* shift  = scale + NHID;                         // [128]

    const int rowBlocks   = NNODES / 16;                  // 3125 (exact)
    const int edgeBlocks  = NEDGES / 8;                   // 100000 (exact)
    const int elemBlocks  = (NNODES * NHID) / 256;        // 25000 (exact)
    const int statBlocks  = (NNODES + 127) / 128;         // 391

    // --- zero scratch accumulators (graph-capture safe)
    hipMemsetAsync(deg, 0, NNODES * sizeof(float), stream);
    hipMemsetAsync(agg, 0, (size_t)NNODES * NHID * sizeof(float), stream);
    hipMemsetAsync(colsum, 0, 2 * NHID * sizeof(float), stream);

    // --- degree / inverse degree
    deg_kernel<<<(NEDGES + 255) / 256, 256, 0, stream>>>(dst, deg);
    invdeg_kernel<<<(NNODES + 255) / 256, 256, 0, stream>>>(deg, invdeg);

    // --- conv1
    scatter_kernel<<<edgeBlocks, 256, 0, stream>>>(src, dst, x, agg);
    sage_gemm_kernel<<<rowBlocks, 256, 0, stream>>>(agg, invdeg, x, W1l, b1l, W1r, h1);
    rownorm_kernel<<<(NNODES * 32 + 255) / 256, 256, 0, stream>>>(h1, 1 /*relu*/);

    // --- batchnorm (batch stats)
    colstats_kernel<<<statBlocks, 128, 0, stream>>>(h1, colsum, colsq);
    bn_finalize_kernel<<<1, 128, 0, stream>>>(colsum, colsq, gamma, beta, scale, shift);
    bn_apply_kernel<<<elemBlocks, 256, 0, stream>>>(h1, scale, shift);

    // --- conv2
    hipMemsetAsync(agg, 0, (size_t)NNODES * NHID * sizeof(float), stream);
    scatter_kernel<<<edgeBlocks, 256, 0, stream>>>(src, dst, h1, agg);
    sage_gemm_kernel<<<rowBlocks, 256, 0, stream>>>(agg, invdeg, h1, W2l, b2l, W2r, h2);
    rownorm_kernel<<<(NNODES * 32 + 255) / 256, 256, 0, stream>>>(h2, 0 /*no relu*/);

    // --- final classifier
    fc_kernel<<<(NNODES * NCLASS + 255) / 256, 256, 0, stream>>>(h2, Wfc, bfc, out);
}